// DistMulti_58471684767977
// MI455X (gfx1250) — compile-verified
//
#include <hip/hip_runtime.h>

typedef __attribute__((ext_vector_type(2))) float v2f;
typedef __attribute__((ext_vector_type(4))) float v4f;
typedef __attribute__((ext_vector_type(8))) float v8f;

#ifndef D_DIM
#define D_DIM 128   // embedding dim, fixed by the reference
#endif

// One wave32 computes a tile of 16 DistMult scores via V_WMMA_F32_16X16X4_F32.
// Custom K-slot mapping (WMMA sums symmetrically over K, so any bijection of
// dims onto (step, slot) works as long as A and B agree):
//   per i in [0,16): lanes 0-15 carry dims 8i..8i+3, lanes 16-31 carry 8i+4..8i+7
//   step 2i   (-> c0): slots {0,1}/{2,3} = first  two floats of each lane's b128
//   step 2i+1 (-> c1): slots {0,1}/{2,3} = second two floats of each lane's b128
// => ONE global_load_b128 per operand feeds TWO WMMA steps (512 B per wave per
// load instruction, every byte consumed exactly once).
// Diagonal of the 16x16 result = the 16 scores.
__global__ __launch_bounds__(256) void distmult_wmma_kernel(
    const float* __restrict__ emb_A,
    const float* __restrict__ emb_B,
    const float* __restrict__ kern,        // rel_kernel[0], D floats
    const int*   __restrict__ edge_pos,    // [2, E] row-major
    const int*   __restrict__ head_batch,  // [E*NEG] flat
    const int*   __restrict__ tail_batch,  // [E*NEG] flat
    float*       __restrict__ out,         // [S] = pos ++ head ++ tail
    int E, int NEG, int S)
{
    const int wave  = blockIdx.x * (blockDim.x >> 5) + (threadIdx.x >> 5);
    const int lane  = threadIdx.x & 31;
    const int nTile = (S + 15) >> 4;
    if (wave >= nTile) return;             // whole-wave exit: EXEC stays all-1s

    const int sBase = wave << 4;
    const int m     = lane & 15;           // score slot within tile (dup in hi half)
    int s = sBase + m;
    if (s >= S) s = S - 1;                 // clamp for a partial last tile

    // Segment-aware index fetch. E and E*NEG are multiples of 16 here, so this
    // branch is wave-uniform in practice (and reconverges regardless).
    const int EH = E * NEG;
    int src, dst;
    if (s < E) {                           // positive edges
        src = edge_pos[s];
        dst = edge_pos[E + s];
    } else if (s < E + EH) {               // head-batch: corrupt heads
        const int t = s - E;
        src = head_batch[t];
        dst = edge_pos[E + t / NEG];
    } else {                               // tail-batch: corrupt tails
        const int t = s - E - EH;
        src = edge_pos[t / NEG];
        dst = tail_batch[t];
    }

    const int half = (lane < 16) ? 0 : 4;  // which 4-dim group this half-wave carries
    const float* __restrict__ arow = emb_A + (size_t)src * D_DIM + half;
    const float* __restrict__ brow = emb_B + (size_t)dst * D_DIM + half;
    const float* __restrict__ krow = kern + half;

    // Start the back half of each 512B row toward the caches early.
    __builtin_prefetch(arow + 64, 0, 0);   // -> global_prefetch_b8
    __builtin_prefetch(brow + 64, 0, 0);

    // Two accumulators pipeline the XDL dependency chain (even/odd steps).
    v8f c0 = {};
    v8f c1 = {};
#pragma unroll
    for (int i = 0; i < D_DIM / 8; ++i) {
        v4f a = *(const v4f*)(arow + 8 * i);
        v4f b = *(const v4f*)(brow + 8 * i);
        v4f w = *(const v4f*)(krow + 8 * i);
        a *= w;                            // fold DistMult relation kernel into A

        const v2f a0 = __builtin_shufflevector(a, a, 0, 1);
        const v2f b0 = __builtin_shufflevector(b, b, 0, 1);
        c0 = __builtin_amdgcn_wmma_f32_16x16x4_f32(
                 false, a0, false, b0, (short)0, c0, false, false);

        const v2f a1 = __builtin_shufflevector(a, a, 2, 3);
        const v2f b1 = __builtin_shufflevector(b, b, 2, 3);
        c1 = __builtin_amdgcn_wmma_f32_16x16x4_f32(
                 false, a1, false, b1, (short)0, c1, false, false);
    }
    const v8f c = c0 + c1;

    // Diagonal extraction: lane l in [0,8) holds score l in c[l];
    // lane l in [24,32) holds score l-16 in c[l-24].
    const int d = (lane < 16) ? lane : (lane - 16);   // target score index m
    float val = c[0];
#pragma unroll
    for (int i = 1; i < 8; ++i) val = ((d & 7) == i) ? c[i] : val;

    const int so = sBase + d;
    if (((lane < 8) || (lane >= 24)) && so < S) out[so] = val;
}

extern "C" void kernel_launch(void* const* d_in, const int* in_sizes, int n_in,
                              void* d_out, int out_size, void* d_ws, size_t ws_size,
                              hipStream_t stream) {
    const float* emb_A      = (const float*)d_in[0];
    const float* emb_B      = (const float*)d_in[1];
    const float* kern       = (const float*)d_in[2];   // rel_kernel, [1, D]
    const int*   edge_pos   = (const int*)d_in[3];     // [2, E]
    const int*   head_batch = (const int*)d_in[4];     // [E, NEG]
    const int*   tail_batch = (const int*)d_in[5];     // [E, NEG]

    const int E   = in_sizes[3] / 2;
    const int NEG = (E > 0) ? (in_sizes[4] / E) : 1;
    const int S   = out_size;                           // E * (1 + 2*NEG)

    const int nTile       = (S + 15) / 16;
    const int wavesPerBlk = 256 / 32;
    const int blocks      = (nTile + wavesPerBlk - 1) / wavesPerBlk;

    distmult_wmma_kernel<<<blocks, 256, 0, stream>>>(
        emb_A, emb_B, kern, edge_pos, head_batch, tail_batch,
        (float*)d_out, E, NEG, S);
}